// DefectAwareAttention_63316407878108
// MI455X (gfx1250) — compile-verified
//
#include <hip/hip_runtime.h>
#include <hip/hip_bf16.h>
#include <math.h>
#include <stdint.h>

typedef __attribute__((ext_vector_type(2))) float v2f;
typedef __attribute__((ext_vector_type(4))) float f4v;
typedef __attribute__((ext_vector_type(8))) float v8f;

#define N_NODES 50000
#define N_EDGES 600000
#define HIDDEN  128
#define HEADS   4
#define HEAD_DIM 32
#define RBF     40
#define LDSPAD  132   // 128 + 4 pad -> conflict-free 16-row fragment reads
#define DEG_CAP 256   // per-node sort capacity (Poisson lambda=12 -> never exceeded)

__device__ __forceinline__ v8f wmma_f32_k4(v2f a, v2f b, v8f c) {
  // D = A(16x4,f32) * B(4x16,f32) + C(16x16,f32)  -- full fp32 matrix pipe
  return __builtin_amdgcn_wmma_f32_16x16x4_f32(
      /*neg_a=*/false, a, /*neg_b=*/false, b,
      /*c_mod=*/(short)0, c, /*reuse_a=*/false, /*reuse_b=*/false);
}

// Async copy 16 bytes global -> LDS (per-lane), tracked by ASYNCcnt.
__device__ __forceinline__ void async_copy16(unsigned lds_byte_addr, const void* gptr) {
  asm volatile("global_load_async_to_lds_b128 %0, %1, off"
               :: "v"(lds_byte_addr), "v"(gptr) : "memory");
}
__device__ __forceinline__ void async_wait0() {
  asm volatile("s_wait_asynccnt 0x0" ::: "memory");
}
__device__ __forceinline__ void ds_wait0() {
  asm volatile("s_wait_dscnt 0x0" ::: "memory");
}

// ---------------------------------------------------------------- init
__global__ void zero_i32(int* __restrict__ p, int n) {
  int i = blockIdx.x * blockDim.x + threadIdx.x;
  if (i < n) p[i] = 0;
}

// ---------------------------------------------------------------- QKV GEMM
// One block = 16 rows of X. A tile staged via async-to-LDS; 8 waves, each
// computes one 16-col tile of Q, K and V (32 fp32 WMMAs each, K=128).
__global__ __launch_bounds__(256)
void qkv_wmma(const float* __restrict__ X,
              const float* __restrict__ Wq, const float* __restrict__ bq,
              const float* __restrict__ Wk, const float* __restrict__ bk,
              const float* __restrict__ Wv, const float* __restrict__ bv,
              float* __restrict__ Q, float* __restrict__ Kout, float* __restrict__ V) {
  __shared__ float As[16 * LDSPAD];
  const int tid  = threadIdx.x;
  const int lane = tid & 31;
  const int wave = tid >> 5;
  const int rowBase = blockIdx.x * 16;

  // async-stage the 16x128 A tile (2 x 16B per thread), padded stride
  const unsigned As_base = (unsigned)(uintptr_t)&As[0];
#pragma unroll
  for (int i = tid; i < 16 * 32; i += 256) {
    int r = i >> 5, c4 = i & 31;
    async_copy16(As_base + (unsigned)(r * LDSPAD + c4 * 4) * 4u,
                 X + (size_t)(rowBase + r) * HIDDEN + c4 * 4);
  }
  async_wait0();
  __syncthreads();

  const int mrow   = lane & 15;
  const int ksel   = (lane >> 4) * 2;   // lanes 0-15: K+0/1, lanes 16-31: K+2/3
  const int rowOff = (lane >> 4) * 8;   // C frag second half holds M+8
  const int col    = wave * 16 + mrow;

  const float* Bm[3] = {Wq, Wk, Wv};
  const float* bm[3] = {bq, bk, bv};
  float*       Om[3] = {Q, Kout, V};

#pragma unroll
  for (int m = 0; m < 3; ++m) {
    const float* B = Bm[m];
    __builtin_prefetch(B + col, 0, 0);  // global_prefetch_b8 on the weight panel
    v8f c = {};
#pragma unroll 8
    for (int kb = 0; kb < HIDDEN; kb += 4) {
      v2f a, b;
      a.x = As[mrow * LDSPAD + kb + ksel];
      a.y = As[mrow * LDSPAD + kb + ksel + 1];
      b.x = B[(size_t)(kb + ksel) * HIDDEN + col];
      b.y = B[(size_t)(kb + ksel + 1) * HIDDEN + col];
      c = wmma_f32_k4(a, b, c);
    }
    float bias = bm[m][col];
    float* O = Om[m];
#pragma unroll
    for (int i = 0; i < 8; ++i)
      O[(size_t)(rowBase + rowOff + i) * HIDDEN + col] = c[i] + bias;
  }
}

// ---------------------------------------------------------------- out GEMM
__global__ __launch_bounds__(256)
void gemm128_wmma(const float* __restrict__ A, const float* __restrict__ B,
                  const float* __restrict__ bias, float* __restrict__ C) {
  __shared__ float As[16 * LDSPAD];
  const int tid  = threadIdx.x;
  const int lane = tid & 31;
  const int wave = tid >> 5;
  const int rowBase = blockIdx.x * 16;

  const unsigned As_base = (unsigned)(uintptr_t)&As[0];
#pragma unroll
  for (int i = tid; i < 16 * 32; i += 256) {
    int r = i >> 5, c4 = i & 31;
    async_copy16(As_base + (unsigned)(r * LDSPAD + c4 * 4) * 4u,
                 A + (size_t)(rowBase + r) * HIDDEN + c4 * 4);
  }
  async_wait0();
  __syncthreads();

  const int mrow   = lane & 15;
  const int ksel   = (lane >> 4) * 2;
  const int rowOff = (lane >> 4) * 8;
  const int col    = wave * 16 + mrow;
  __builtin_prefetch(B + col, 0, 0);

  v8f c = {};
#pragma unroll 8
  for (int kb = 0; kb < HIDDEN; kb += 4) {
    v2f a, b;
    a.x = As[mrow * LDSPAD + kb + ksel];
    a.y = As[mrow * LDSPAD + kb + ksel + 1];
    b.x = B[(size_t)(kb + ksel) * HIDDEN + col];
    b.y = B[(size_t)(kb + ksel + 1) * HIDDEN + col];
    c = wmma_f32_k4(a, b, c);
  }
  float bv = bias[col];
#pragma unroll
  for (int i = 0; i < 8; ++i)
    C[(size_t)(rowBase + rowOff + i) * HIDDEN + col] = c[i] + bv;
}

// ---------------------------------------------------------------- edge scores
// One wave per 16 edges. WMMA MLP: hidden = silu(rbf@Wg1+bg1) [16x128],
// geo = hidden@Wg2+bg2 [16x4 padded to 16]. Then per-lane Q.K dots
// (lane = edge, half-wave picks head pair) + defect bias; stores score[E,4].
__global__ __launch_bounds__(32)
void edge_score(const float* __restrict__ rbf, const int* __restrict__ ei,
                const int* __restrict__ isdef,
                const float* __restrict__ Wg1, const float* __restrict__ bg1,
                const float* __restrict__ Wg2, const float* __restrict__ bg2,
                const float* __restrict__ dbias,
                const float* __restrict__ Q, const float* __restrict__ Kmat,
                float* __restrict__ score) {
  __shared__ float hid[16 * LDSPAD];
  __shared__ float geo[16][HEADS];
  const int lane = threadIdx.x & 31;
  const int e0   = blockIdx.x * 16;
  const int mrow = lane & 15;
  const int ksel   = (lane >> 4) * 2;
  const int rowOff = (lane >> 4) * 8;

  // hidden = silu(rbf @ Wg1 + bg1)
#pragma unroll
  for (int nt = 0; nt < 8; ++nt) {
    const int col = nt * 16 + mrow;
    v8f c = {};
#pragma unroll
    for (int kb = 0; kb < RBF; kb += 4) {
      v2f a, b;
      a.x = rbf[(size_t)(e0 + mrow) * RBF + kb + ksel];
      a.y = rbf[(size_t)(e0 + mrow) * RBF + kb + ksel + 1];
      b.x = Wg1[(size_t)(kb + ksel) * HIDDEN + col];
      b.y = Wg1[(size_t)(kb + ksel + 1) * HIDDEN + col];
      c = wmma_f32_k4(a, b, c);
    }
    float bb = bg1[col];
#pragma unroll
    for (int i = 0; i < 8; ++i) {
      float x = c[i] + bb;
      hid[(rowOff + i) * LDSPAD + col] = x / (1.0f + __expf(-x));  // silu
    }
  }
  __syncthreads();

  // geo = hidden @ Wg2 + bg2 (columns >= HEADS zero-padded)
  {
    v8f c = {};
#pragma unroll 8
    for (int kb = 0; kb < HIDDEN; kb += 4) {
      v2f a, b;
      a.x = hid[mrow * LDSPAD + kb + ksel];
      a.y = hid[mrow * LDSPAD + kb + ksel + 1];
      b.x = (mrow < HEADS) ? Wg2[(size_t)(kb + ksel) * HEADS + mrow] : 0.0f;
      b.y = (mrow < HEADS) ? Wg2[(size_t)(kb + ksel + 1) * HEADS + mrow] : 0.0f;
      c = wmma_f32_k4(a, b, c);
    }
    if (mrow < HEADS) {
      float bb = bg2[mrow];
#pragma unroll
      for (int i = 0; i < 8; ++i) geo[rowOff + i][mrow] = c[i] + bb;
    }
  }
  __syncthreads();

  // per-lane scores: edge = e0+mrow, this half-wave handles heads h0,h0+1
  const int e   = e0 + mrow;
  const int h0  = (lane >> 4) * 2;
  const int src = ei[e];
  const int dst = ei[N_EDGES + e];
  const int code = isdef[src] * 2 + isdef[dst];
  const float inv_sqrt = 0.1767766952966369f;  // 1/sqrt(32)

#pragma unroll
  for (int hh = 0; hh < 2; ++hh) {
    const int h = h0 + hh;
    const f4v* q = (const f4v*)(Q    + (size_t)src * HIDDEN + h * HEAD_DIM);
    const f4v* k = (const f4v*)(Kmat + (size_t)dst * HIDDEN + h * HEAD_DIM);
    float acc = 0.0f;
#pragma unroll
    for (int d = 0; d < 8; ++d) {
      f4v qa = q[d], ka = k[d];
      acc += qa.x * ka.x + qa.y * ka.y + qa.z * ka.z + qa.w * ka.w;
    }
    score[(size_t)e * HEADS + h] = acc * inv_sqrt + geo[mrow][h] + dbias[h * 4 + code];
  }
}

// ---------------------------------------------------------------- CSR build
__global__ __launch_bounds__(256)
void degree_count(const int* __restrict__ ei, int* __restrict__ deg) {
  int e = blockIdx.x * blockDim.x + threadIdx.x;
  if (e < N_EDGES) atomicAdd(&deg[ei[N_EDGES + e]], 1);
}

// single-block exclusive scan of 50000 degrees -> offsets[N+1], cursor copy
__global__ __launch_bounds__(1024)
void scan_offsets(const int* __restrict__ deg, int* __restrict__ offsets,
                  int* __restrict__ cursor) {
  __shared__ int buf[1024];
  __shared__ int carry;
  const int tid = threadIdx.x;
  if (tid == 0) carry = 0;
  __syncthreads();
  for (int base = 0; base < N_NODES; base += 1024) {
    const int i = base + tid;
    const int v = (i < N_NODES) ? deg[i] : 0;
    buf[tid] = v;
    __syncthreads();
#pragma unroll
    for (int off = 1; off < 1024; off <<= 1) {
      int t = (tid >= off) ? buf[tid - off] : 0;
      __syncthreads();
      buf[tid] += t;
      __syncthreads();
    }
    const int incl = buf[tid];
    const int c0 = carry;
    if (i < N_NODES) {
      int excl = c0 + incl - v;
      offsets[i] = excl;
      cursor[i] = excl;
    }
    __syncthreads();
    if (tid == 1023) carry = c0 + incl;
    __syncthreads();
  }
  if (tid == 0) offsets[N_NODES] = carry;   // == N_EDGES
}

__global__ __launch_bounds__(256)
void scatter_edges(const int* __restrict__ ei, int* __restrict__ cursor,
                   int* __restrict__ edge_ids) {
  int e = blockIdx.x * blockDim.x + threadIdx.x;
  if (e < N_EDGES) {
    int p = atomicAdd(&cursor[ei[N_EDGES + e]], 1);
    edge_ids[p] = e;
  }
}

// ---------------------------------------------------------------- per-node softmax reduce
// One wave per destination node (8 nodes / 256-thread block). Lane owns 4
// hidden dims (head = lane>>3). Segment is rank-sorted by edge id in LDS so
// the FP summation order is identical on every call (deterministic), then
// max / exp-sum / weighted V accumulation -- no FP atomics anywhere.
__global__ __launch_bounds__(256)
void node_reduce(const int* __restrict__ offsets, const int* __restrict__ edge_ids,
                 const int* __restrict__ ei, const float* __restrict__ score,
                 const float* __restrict__ V, float* __restrict__ accum) {
  __shared__ int sorted[8][DEG_CAP];
  const int wave = threadIdx.x >> 5;
  const int lane = threadIdx.x & 31;
  const int n    = blockIdx.x * 8 + wave;
  const int beg  = offsets[n];
  const int deg  = offsets[n + 1] - beg;
  const int h    = lane >> 3;      // head for this lane's dims
  const int d4   = lane * 4;       // hidden dim base (float4)

  const bool use_sorted = (deg <= DEG_CAP);
  if (use_sorted) {
    for (int base = 0; base < deg; base += 32) {
      int l = base + lane;
      if (l < deg) {
        int myid = edge_ids[beg + l];
        int rank = 0;
        for (int j = 0; j < deg; ++j) rank += (edge_ids[beg + j] < myid);
        sorted[wave][rank] = myid;
      }
    }
  }
  ds_wait0();  // own-wave LDS visibility (DS ops in-order per wave)

  f4v acc = {0.f, 0.f, 0.f, 0.f};
  float den = 0.f;
  float m = -3.402823466e38f;
  for (int j = 0; j < deg; ++j) {
    int e = use_sorted ? sorted[wave][j] : edge_ids[beg + j];
    m = fmaxf(m, score[(size_t)e * HEADS + h]);
  }
  for (int j = 0; j < deg; ++j) {
    int e = use_sorted ? sorted[wave][j] : edge_ids[beg + j];
    float ex = __expf(score[(size_t)e * HEADS + h] - m);
    den += ex;
    f4v v = *(const f4v*)(V + (size_t)ei[e] * HIDDEN + d4);  // ei[e] == src
    acc.x += v.x * ex; acc.y += v.y * ex; acc.z += v.z * ex; acc.w += v.w * ex;
  }
  f4v outv = {0.f, 0.f, 0.f, 0.f};
  if (den > 0.f) {
    float r = 1.0f / den;
    outv.x = acc.x * r; outv.y = acc.y * r; outv.z = acc.z * r; outv.w = acc.w * r;
  }
  *(f4v*)(accum + (size_t)n * HIDDEN + d4) = outv;
}

// ---------------------------------------------------------------- launch
extern "C" void kernel_launch(void* const* d_in, const int* in_sizes, int n_in,
                              void* d_out, int out_size, void* d_ws, size_t ws_size,
                              hipStream_t stream) {
  const float* x     = (const float*)d_in[0];
  const int*   ei    = (const int*)  d_in[1];
  const float* rbf   = (const float*)d_in[2];
  const int*   isdef = (const int*)  d_in[3];
  const float* Wq = (const float*)d_in[4],  *bq = (const float*)d_in[5];
  const float* Wk = (const float*)d_in[6],  *bk = (const float*)d_in[7];
  const float* Wv = (const float*)d_in[8],  *bv = (const float*)d_in[9];
  const float* Wo = (const float*)d_in[10], *bo = (const float*)d_in[11];
  const float* Wg1 = (const float*)d_in[12], *bg1 = (const float*)d_in[13];
  const float* Wg2 = (const float*)d_in[14], *bg2 = (const float*)d_in[15];
  const float* db  = (const float*)d_in[16];

  float* ws = (float*)d_ws;
  const size_t NH = (size_t)N_NODES * HIDDEN;            // 6,400,000
  float* Q     = ws;                                     // [N,128]
  float* Km    = ws + NH;                                // [N,128]
  float* Vm    = ws + 2 * NH;                            // [N,128]
  float* score = ws + 3 * NH;                            // [E,4]
  float* accum = score + (size_t)N_EDGES * HEADS;        // [N,128]
  int* deg      = (int*)(accum + NH);                    // [N]
  int* offsets  = deg + N_NODES;                         // [N+1]
  int* cursor   = offsets + (N_NODES + 1);               // [N]
  int* edge_ids = cursor + N_NODES;                      // [E]

  // 1) reset degree histogram (graph-replay safe; everything else fully rewritten)
  zero_i32<<<(N_NODES + 255) / 256, 256, 0, stream>>>(deg, N_NODES);

  // 2) Q/K/V projections (WMMA f32, async-LDS staged)
  qkv_wmma<<<N_NODES / 16, 256, 0, stream>>>(x, Wq, bq, Wk, bk, Wv, bv, Q, Km, Vm);

  // 3) per-edge scores: WMMA geo-MLP + Q.K + defect bias
  edge_score<<<N_EDGES / 16, 32, 0, stream>>>(rbf, ei, isdef, Wg1, bg1, Wg2, bg2,
                                              db, Q, Km, score);

  // 4) CSR by destination: histogram -> scan -> scatter
  degree_count<<<(N_EDGES + 255) / 256, 256, 0, stream>>>(ei, deg);
  scan_offsets<<<1, 1024, 0, stream>>>(deg, offsets, cursor);
  scatter_edges<<<(N_EDGES + 255) / 256, 256, 0, stream>>>(ei, cursor, edge_ids);

  // 5) deterministic per-node segment softmax + message aggregation (no FP atomics)
  node_reduce<<<N_NODES / 8, 256, 0, stream>>>(offsets, edge_ids, ei, score, Vm, accum);

  // 6) output projection (WMMA f32) straight into d_out
  gemm128_wmma<<<N_NODES / 16, 256, 0, stream>>>(accum, Wo, bo, (float*)d_out);
}